// InputModuleLayer_5085241278807
// MI455X (gfx1250) — compile-verified
//
#include <hip/hip_runtime.h>
#include <hip/hip_bf16.h>
#include <stdint.h>

// Problem constants from the reference (B=16, L=2048, D=1024).
#define LROW   2048            // tokens per batch row
#define DDIM   1024            // floats per token row
#define ROWB   4096            // bytes per token row (4 KiB)
#define T1     256             // threads, phase 1
#define PER    8               // tokens per thread, phase 1 (256*8 = 2048)
#define T2     256             // threads, phase 2 (256 lanes * 16 B = 4 KiB row)
#define ROWS   4               // rows copied per block, phase 2 (pipelining depth)

// ---------------------------------------------------------------------------
// Phase 1: per batch row, compute the stable-partition permutation.
// Writes src_for_dst[b*L + dest] = l  (inverse permutation, scatter form),
// so phase 2 can do a pure gather: out[b, j, :] = sentout[b, perm[b*L+j], :].
// ---------------------------------------------------------------------------
__global__ void build_perm_kernel(const int* __restrict__ nl_input,
                                  int* __restrict__ perm) {
    const int b = blockIdx.x;
    const int t = threadIdx.x;
    const int* row = nl_input + (size_t)b * LROW;

    __shared__ int s_scan[T1];

    const int base = t * PER;
    int flags[PER];
    int c = 0;
#pragma unroll
    for (int i = 0; i < PER; ++i) {
        flags[i] = (row[base + i] != 0) ? 1 : 0;
        c += flags[i];
    }
    s_scan[t] = c;
    __syncthreads();

    // Hillis-Steele inclusive scan over 256 per-thread counts.
#pragma unroll
    for (int off = 1; off < T1; off <<= 1) {
        int v = (t >= off) ? s_scan[t - off] : 0;
        __syncthreads();
        s_scan[t] += v;
        __syncthreads();
    }

    const int total_nonpad = s_scan[T1 - 1];
    int nb = s_scan[t] - c;   // nonpads strictly before my first element

#pragma unroll
    for (int i = 0; i < PER; ++i) {
        const int l = base + i;
        // pads_before(l) = l - nb  =>  stable partition destination:
        const int dest = flags[i] ? nb : (total_nonpad + (l - nb));
        perm[(size_t)b * LROW + dest] = l;
        nb += flags[i];
    }
}

// ---------------------------------------------------------------------------
// Phase 2: permuted row copy, ROWS output rows (4 KiB each) per block.
// CDNA5 async data path (ASYNCcnt-tracked):
//   * issue ROWS GLOBAL_LOAD_ASYNC_TO_LDS_B128 per lane (deep in-flight read
//     window: ROWS * 512 B per wave) ...
//   * ... one s_wait_asynccnt 0 ...
//   * ... then ROWS GLOBAL_STORE_ASYNC_FROM_LDS_B128 back-to-back, drained by
//     s_endpgm's implicit wait-idle.
// Each lane stages its own private 16 B LDS slots, so no barrier is needed.
// ---------------------------------------------------------------------------
__global__ void gather_rows_async_kernel(const float* __restrict__ src,
                                         const int* __restrict__ perm,
                                         float* __restrict__ dst) {
    __shared__ __align__(16) char buf[ROWS * ROWB];   // 16 KiB staging

    const int row0 = blockIdx.x * ROWS;   // global output row = b*LROW + j
    const unsigned tid = threadIdx.x;

    unsigned long long ga[ROWS], gd[ROWS];
    unsigned lds[ROWS];

#pragma unroll
    for (int r = 0; r < ROWS; ++r) {
        const int row = row0 + r;
        const int b  = row >> 11;            // LROW = 2048
        const int sl = perm[row];            // source token index in batch b
        ga[r] = (unsigned long long)(uintptr_t)
                ((const char*)src + (((size_t)b << 11) + (size_t)sl) * ROWB
                                  + (size_t)tid * 16);
        gd[r] = (unsigned long long)(uintptr_t)
                ((char*)dst + (size_t)row * ROWB + (size_t)tid * 16);
        // Low 32 bits of a flat shared pointer == LDS byte offset.
        lds[r] = (unsigned)(uintptr_t)(&buf[r * ROWB + tid * 16]);
    }

    // Fill the read pipeline: ROWS async B128 loads in flight per lane.
#pragma unroll
    for (int r = 0; r < ROWS; ++r)
        asm volatile("global_load_async_to_lds_b128 %0, %1, off"
                     :: "v"(lds[r]), "v"(ga[r]) : "memory");

    // Single wait: all of this wave's loads have landed in LDS.
    asm volatile("s_wait_asynccnt 0x0" ::: "memory");

    // Drain writes with no stalls; s_endpgm's implicit wait-idle covers them.
#pragma unroll
    for (int r = 0; r < ROWS; ++r)
        asm volatile("global_store_async_from_lds_b128 %0, %1, off"
                     :: "v"(gd[r]), "v"(lds[r]) : "memory");
}

// ---------------------------------------------------------------------------
// Launch
// ---------------------------------------------------------------------------
extern "C" void kernel_launch(void* const* d_in, const int* in_sizes, int n_in,
                              void* d_out, int out_size, void* d_ws, size_t ws_size,
                              hipStream_t stream) {
    const float* sentout = (const float*)d_in[0];   // [B, L, D] f32
    const int*   nl      = (const int*)d_in[1];     // [B, L] ids (0 = pad)
    float*       out     = (float*)d_out;           // [B, L, D] f32

    const int BL = in_sizes[1];          // B * L
    const int B  = BL / LROW;

    int* perm = (int*)d_ws;              // BL ints of scratch (128 KiB)

    build_perm_kernel<<<B, T1, 0, stream>>>(nl, perm);
    gather_rows_async_kernel<<<BL / ROWS, T2, 0, stream>>>(sentout, perm, out);
}